// CBFBarrierNetLayer_14766097563746
// MI455X (gfx1250) — compile-verified
//
#include <hip/hip_runtime.h>

// Batched tiny-QP (CBF safety filter) solver: one QP per lane, fully in registers.
// n = 2 controls, m = 7 constraints (3 data CBF rows + 4 constant box rows).
// Dual projected-gradient (<=300 iters with bit-exact stationarity early-exit)
// + active-set KKT polish (branchless pivoted 7x7 Gaussian elimination),
// mirroring the reference algorithm. Deliberately VALU-only: per-QP 7x2 ops
// with a serial max(.,0) recurrence cannot utilize WMMA (<3% MAC utilization
// best case, and only on the ~1%-of-FLOPs setup stage).

#define QP_N_ITERS 300     // must stay divisible by the block size below
#define QP_BLK 4           // early-exit check granularity (300 = 4*75)
#define QP_ACT_TOL 1e-5f

__global__ __launch_bounds__(256) void cbf_qp_kernel(
    const float* __restrict__ u_ref,   // [B, 2]
    const float* __restrict__ A_cbf,   // [B, 3, 2]
    const float* __restrict__ b_cbf,   // [B, 3]
    float* __restrict__ out,           // [B, 2]
    int B)
{
  const int b = blockIdx.x * blockDim.x + threadIdx.x;
  if (b >= B) return;

  // ---- Loads (8-byte vectorized where stride permits) ----
  const float2 ur2 = reinterpret_cast<const float2*>(u_ref)[b];
  const float urx = ur2.x, ury = ur2.y;
  const float2* A2 = reinterpret_cast<const float2*>(A_cbf) + 3 * b;
  const float2 a0 = A2[0], a1 = A2[1], a2 = A2[2];
  const float bc0 = b_cbf[3 * b + 0];
  const float bc1 = b_cbf[3 * b + 1];
  const float bc2 = b_cbf[3 * b + 2];

  const float ax0 = a0.x, ay0 = a0.y;
  const float ax1 = a1.x, ay1 = a1.y;
  const float ax2 = a2.x, ay2 = a2.y;

  // ---- q = G u_ref - h (box rows specialized: h_box = [1,.5,1,.5]) ----
  const float q0 = fmaf(ax0, urx, fmaf(ay0, ury, -bc0));
  const float q1 = fmaf(ax1, urx, fmaf(ay1, ury, -bc1));
  const float q2 = fmaf(ax2, urx, fmaf(ay2, ury, -bc2));
  const float q3 =  urx - 1.0f;
  const float q4 =  ury - 0.5f;
  const float q5 = -urx - 1.0f;
  const float q6 = -ury - 0.5f;

  // ---- Lipschitz bound: L = max_i sum_j |M_ij|, M = G G^T ----
  // CBF row i: sum_j |a_i.a_j| + 2|ax_i| + 2|ay_i|
  // Box rows (3&5): sum_j |ax_j| + 2 ; (4&6): sum_j |ay_j| + 2
  const float d00 = fmaf(ax0, ax0, ay0 * ay0);
  const float d11 = fmaf(ax1, ax1, ay1 * ay1);
  const float d22 = fmaf(ax2, ax2, ay2 * ay2);
  const float d01 = fabsf(fmaf(ax0, ax1, ay0 * ay1));
  const float d02 = fabsf(fmaf(ax0, ax2, ay0 * ay2));
  const float d12 = fabsf(fmaf(ax1, ax2, ay1 * ay2));
  const float r0 = d00 + d01 + d02 + 2.0f * (fabsf(ax0) + fabsf(ay0));
  const float r1 = d01 + d11 + d12 + 2.0f * (fabsf(ax1) + fabsf(ay1));
  const float r2 = d02 + d12 + d22 + 2.0f * (fabsf(ax2) + fabsf(ay2));
  const float rbx = fabsf(ax0) + fabsf(ax1) + fabsf(ax2) + 2.0f;
  const float rby = fabsf(ay0) + fabsf(ay1) + fabsf(ay2) + 2.0f;
  const float L = fmaxf(fmaxf(fmaxf(r0, r1), fmaxf(r2, rbx)), rby);
  const float step = 1.0f / (L + 1e-6f);

  // ---- Dual PGD: lam = max(lam + step*(q - G(G^T lam)), 0) ----
  const float sq0 = step * q0, sq1 = step * q1, sq2 = step * q2;
  const float sq3 = step * q3, sq4 = step * q4;
  const float sq5 = step * q5, sq6 = step * q6;
  // Pre-scaled row coefficients: keeps step*wx/step*wy off the critical path
  // of the three data-row FMA updates.
  const float sax0 = step * ax0, say0 = step * ay0;
  const float sax1 = step * ax1, say1 = step * ay1;
  const float sax2 = step * ax2, say2 = step * ay2;

  float l0 = 0.f, l1 = 0.f, l2 = 0.f, l3 = 0.f, l4 = 0.f, l5 = 0.f, l6 = 0.f;

  // Early exit is bit-exact: the update is a pure map f(lam). If a 4-iteration
  // block is the identity on lam (bitwise), then — since the remaining
  // iteration count is a multiple of 4 (300 = 4*75) — the remaining blocks are
  // also the identity, so lam_300 equals the current lam exactly (this also
  // covers ulp-level limit cycles whose period divides 4). Wave-uniform branch.
  for (int blk = 0; blk < QP_N_ITERS / QP_BLK; ++blk) {
    const float p0 = l0, p1 = l1, p2 = l2, p3 = l3,
                p4 = l4, p5 = l5, p6 = l6;
#pragma unroll
    for (int s = 0; s < QP_BLK; ++s) {
      // w = G^T lam ; box rows contribute (l3-l5, l4-l6)
      const float wx = fmaf(ax2, l2, fmaf(ax1, l1, fmaf(ax0, l0, l3 - l5)));
      const float wy = fmaf(ay2, l2, fmaf(ay1, l1, fmaf(ay0, l0, l4 - l6)));
      const float swx = step * wx;   // only feeds box rows
      const float swy = step * wy;
      l0 = fmaxf(fmaf(-sax0, wx, fmaf(-say0, wy, l0 + sq0)), 0.f);
      l1 = fmaxf(fmaf(-sax1, wx, fmaf(-say1, wy, l1 + sq1)), 0.f);
      l2 = fmaxf(fmaf(-sax2, wx, fmaf(-say2, wy, l2 + sq2)), 0.f);
      l3 = fmaxf((l3 + sq3) - swx, 0.f);
      l4 = fmaxf((l4 + sq4) - swy, 0.f);
      l5 = fmaxf((l5 + sq5) + swx, 0.f);
      l6 = fmaxf((l6 + sq6) + swy, 0.f);
    }
    const bool changed = (p0 != l0) | (p1 != l1) | (p2 != l2) | (p3 != l3) |
                         (p4 != l4) | (p5 != l5) | (p6 != l6);
    if (!__any(changed)) break;
  }

  // ---- Active-set polish ----
  float Gx[7], Gy[7], q[7], lam[7];
  Gx[0] = ax0;  Gy[0] = ay0;  q[0] = q0;  lam[0] = l0;
  Gx[1] = ax1;  Gy[1] = ay1;  q[1] = q1;  lam[1] = l1;
  Gx[2] = ax2;  Gy[2] = ay2;  q[2] = q2;  lam[2] = l2;
  Gx[3] =  1.f; Gy[3] =  0.f; q[3] = q3;  lam[3] = l3;
  Gx[4] =  0.f; Gy[4] =  1.f; q[4] = q4;  lam[4] = l4;
  Gx[5] = -1.f; Gy[5] =  0.f; q[5] = q5;  lam[5] = l5;
  Gx[6] =  0.f; Gy[6] = -1.f; q[6] = q6;  lam[6] = l6;

  float act[7];
#pragma unroll
  for (int i = 0; i < 7; ++i) act[i] = (lam[i] > QP_ACT_TOL) ? 1.f : 0.f;

  // Augmented system: Msys = M.*(act act^T) + diag(1-act) + 1e-8 I ; rhs = act.*q
  float Asys[7][8];
#pragma unroll
  for (int i = 0; i < 7; ++i) {
#pragma unroll
    for (int j = 0; j < 7; ++j) {
      const float Mij = fmaf(Gx[i], Gx[j], Gy[i] * Gy[j]);
      float v = Mij * (act[i] * act[j]);
      if (i == j) v += (1.f - act[i]) + 1e-8f;
      Asys[i][j] = v;
    }
    Asys[i][7] = act[i] * q[i];
  }

  // Gaussian elimination with partial pivoting via fully-unrolled branchless
  // conditional row swaps: all indices compile-time constant -> stays in VGPRs.
#pragma unroll
  for (int k = 0; k < 7; ++k) {
#pragma unroll
    for (int j = k + 1; j < 7; ++j) {
      const bool sw = fabsf(Asys[j][k]) > fabsf(Asys[k][k]);
#pragma unroll
      for (int c = k; c < 8; ++c) {
        const float t0 = Asys[k][c];
        const float t1 = Asys[j][c];
        Asys[k][c] = sw ? t1 : t0;
        Asys[j][c] = sw ? t0 : t1;
      }
    }
    const float inv = 1.0f / Asys[k][k];
#pragma unroll
    for (int j = k + 1; j < 7; ++j) {
      const float f = Asys[j][k] * inv;
#pragma unroll
      for (int c = k + 1; c < 8; ++c)
        Asys[j][c] = fmaf(-f, Asys[k][c], Asys[j][c]);
    }
  }

  float x[7];
#pragma unroll
  for (int k = 6; k >= 0; --k) {
    float s = Asys[k][7];
#pragma unroll
    for (int c = k + 1; c < 7; ++c)
      s = fmaf(-Asys[k][c], x[c], s);
    x[k] = s / Asys[k][k];
  }

  // lam = max(x,0)*act ; u = u_ref - G^T lam
  float lx = 0.f, ly = 0.f;
#pragma unroll
  for (int i = 0; i < 7; ++i) {
    const float lf = fmaxf(x[i], 0.f) * act[i];
    lx = fmaf(Gx[i], lf, lx);
    ly = fmaf(Gy[i], lf, ly);
  }

  reinterpret_cast<float2*>(out)[b] = make_float2(urx - lx, ury - ly);
}

extern "C" void kernel_launch(void* const* d_in, const int* in_sizes, int n_in,
                              void* d_out, int out_size, void* d_ws, size_t ws_size,
                              hipStream_t stream) {
  const float* u_ref = (const float*)d_in[0];   // [B,2,1] f32
  const float* A_cbf = (const float*)d_in[1];   // [B,3,2] f32
  const float* b_cbf = (const float*)d_in[2];   // [B,3,1] f32
  float* out = (float*)d_out;                    // [B,2,1] f32

  const int B = in_sizes[0] / 2;
  const int threads = 256;                       // 8 wave32 waves per block
  const int blocks = (B + threads - 1) / threads;

  hipLaunchKernelGGL(cbf_qp_kernel, dim3(blocks), dim3(threads), 0, stream,
                     u_ref, A_cbf, b_cbf, out, B);
}